// RnCLoss_5463198400819
// MI455X (gfx1250) — compile-verified
//
#include <hip/hip_runtime.h>
#include <math.h>

#define N 512
#define D 256
#define INV_T 10.0f                  // 1 / TEMPERATURE
#define F32_EPS 1.1920928955078125e-07f

typedef __attribute__((ext_vector_type(2))) float v2f;
typedef __attribute__((ext_vector_type(8))) float v8f;

// ---------------- Kernel 1: zero the epsilon flag ----------------
__global__ void rnc_init(unsigned* flag) {
    if (threadIdx.x == 0) *flag = 0u;
}

// ---------------- Kernel 2: Gram matrix G = X * X^T via f32 WMMA ----------------
// One wave per 16x16 output tile; 8 waves (256 threads) per block; 1024 tiles total.
// A-fragment layout (16x4 f32): lanes 0-15 -> M=lane, K=kk+{0,1}; lanes 16-31 -> M=lane-16, K=kk+{2,3}.
// B = X^T, so B-fragment for column block tn loads X rows tn+0..15 with the same lane pattern.
__global__ __launch_bounds__(256) void gram_wmma(const float* __restrict__ X,
                                                 float* __restrict__ G) {
    const int wave = threadIdx.x >> 5;
    const int lane = threadIdx.x & 31;
    const int tile = blockIdx.x * 8 + wave;      // 0..1023
    const int tm = (tile >> 5) << 4;             // row-tile base
    const int tn = (tile & 31) << 4;             // col-tile base
    const int r  = lane & 15;
    const int khalf = (lane >> 4) << 1;          // 0 for lanes 0-15, 2 for lanes 16-31

    const float* __restrict__ arow = X + (tm + r) * D + khalf;
    const float* __restrict__ brow = X + (tn + r) * D + khalf;

    v8f c = {};
    #pragma unroll 4
    for (int kk = 0; kk < D; kk += 4) {
        v2f a = *(const v2f*)(arow + kk);
        v2f b = *(const v2f*)(brow + kk);
        // 8 args: (neg_a, A, neg_b, B, c_mod, C, reuse_a, reuse_b)
        c = __builtin_amdgcn_wmma_f32_16x16x4_f32(false, a, false, b,
                                                  (short)0, c, false, false);
    }

    // C/D layout: VGPR i, lanes 0-15 -> M=i, N=lane; lanes 16-31 -> M=i+8, N=lane-16.
    const int mbase = tm + ((lane >> 4) << 3);
    const int ncol  = tn + r;
    #pragma unroll
    for (int i = 0; i < 8; ++i)
        G[(mbase + i) * N + ncol] = c[i];
}

// ---------------- Kernel 3: global any(exp_fs == 0) check ----------------
__global__ __launch_bounds__(256) void underflow_flag(const float* __restrict__ G,
                                                      unsigned* __restrict__ flag) {
    __shared__ float fsrow[N];
    __shared__ float red[256];
    const int i = blockIdx.x;
    const float ni = G[i * N + i];

    float lmax = -INFINITY;
    for (int j = threadIdx.x; j < N; j += 256) {
        float fs;
        if (j == i) {
            fs = 0.0f;
        } else {
            float d2 = fmaxf(ni + G[j * N + j] - 2.0f * G[i * N + j], 0.0f);
            fs = -sqrtf(d2) * INV_T;
        }
        fsrow[j] = fs;
        lmax = fmaxf(lmax, fs);
    }
    red[threadIdx.x] = lmax;
    __syncthreads();
    for (int s = 128; s > 0; s >>= 1) {
        if (threadIdx.x < s) red[threadIdx.x] = fmaxf(red[threadIdx.x], red[threadIdx.x + s]);
        __syncthreads();
    }
    const float m = red[0];

    bool anyzero = false;
    for (int j = threadIdx.x; j < N; j += 256)
        anyzero |= (expf(fsrow[j] - m) == 0.0f);
    if (anyzero) atomicOr(flag, 1u);   // idempotent -> deterministic
}

// ---------------- Kernel 4: per-row masked softmax denominator + partial loss ----------------
// One block (8 wave32) per row i. (ld, exp) packed as float2 in LDS -> one ds_load_b64
// per k; each thread fuses its two j's (tid, tid+256) into a single k-pass.
__global__ __launch_bounds__(256) void row_loss_kernel(const float* __restrict__ G,
                                                       const float* __restrict__ Tg,
                                                       const unsigned* __restrict__ flag,
                                                       float* __restrict__ row_loss) {
    __shared__ float  fsrow[N];
    __shared__ float2 lerow[N];      // .x = label dist, .y = exp(fs - m) + eps
    __shared__ float  red[256];
    const int tid = threadIdx.x;
    const int i = blockIdx.x;
    const float ni = G[i * N + i];
    const float ti = Tg[i];

    float lmax = -INFINITY;
    for (int j = tid; j < N; j += 256) {
        float fs, ld;
        if (j == i) {
            fs = 0.0f; ld = 0.0f;
        } else {
            float d2 = fmaxf(ni + G[j * N + j] - 2.0f * G[i * N + j], 0.0f);
            fs = -sqrtf(d2) * INV_T;
            ld = fabsf(ti - Tg[j]);
        }
        fsrow[j] = fs;
        lerow[j].x = ld;
        lmax = fmaxf(lmax, fs);
    }
    red[tid] = lmax;
    __syncthreads();
    for (int s = 128; s > 0; s >>= 1) {
        if (tid < s) red[tid] = fmaxf(red[tid], red[tid + s]);
        __syncthreads();
    }
    const float m = red[0];
    const float eps = (*flag) ? F32_EPS : 0.0f;

    for (int j = tid; j < N; j += 256)
        lerow[j].y = expf(fsrow[j] - m) + eps;
    __syncthreads();

    const int j0 = tid;
    const int j1 = tid + 256;
    const float ldj0 = lerow[j0].x;
    const float ldj1 = lerow[j1].x;
    float den0 = 0.0f, den1 = 0.0f;
    #pragma unroll 4
    for (int k = 0; k < N; ++k) {
        const float2 le = lerow[k];
        const float e = (k != i) ? le.y : 0.0f;   // off-diagonal k only
        den0 += (le.x >= ldj0) ? e : 0.0f;        // mask: ld[i,k] >= ld[i,j]
        den1 += (le.x >= ldj1) ? e : 0.0f;
    }
    float acc = 0.0f;
    if (j0 != i) acc += (fsrow[j0] - m) - logf(den0);
    if (j1 != i) acc += (fsrow[j1] - m) - logf(den1);

    red[tid] = acc;
    __syncthreads();
    for (int s = 128; s > 0; s >>= 1) {
        if (tid < s) red[tid] += red[tid + s];
        __syncthreads();
    }
    if (tid == 0) row_loss[i] = red[0];
}

// ---------------- Kernel 5: deterministic final reduction ----------------
__global__ void final_reduce(const float* __restrict__ row_loss, float* __restrict__ out) {
    if (threadIdx.x == 0 && blockIdx.x == 0) {
        float s = 0.0f;
        for (int i = 0; i < N; ++i) s += row_loss[i];   // fixed order
        out[0] = -s / (float)(N * (N - 1));
    }
}

extern "C" void kernel_launch(void* const* d_in, const int* in_sizes, int n_in,
                              void* d_out, int out_size, void* d_ws, size_t ws_size,
                              hipStream_t stream) {
    (void)in_sizes; (void)n_in; (void)out_size; (void)ws_size;
    const float* X  = (const float*)d_in[0];   // [N, D] fp32
    const float* Tg = (const float*)d_in[1];   // [N, 1] fp32
    float* out = (float*)d_out;                // scalar loss

    char* ws = (char*)d_ws;
    float*    G        = (float*)ws;                                  // N*N fp32 = 1 MB
    unsigned* flag     = (unsigned*)(ws + (size_t)N * N * sizeof(float));
    float*    row_loss = (float*)(ws + (size_t)N * N * sizeof(float) + 256);

    rnc_init<<<1, 64, 0, stream>>>(flag);
    gram_wmma<<<(N / 16) * (N / 16) / 8, 256, 0, stream>>>(X, G);     // 128 blocks x 8 waves
    underflow_flag<<<N, 256, 0, stream>>>(G, flag);
    row_loss_kernel<<<N, 256, 0, stream>>>(G, Tg, flag, row_loss);
    final_reduce<<<1, 32, 0, stream>>>(row_loss, out);
}